// DegeneratePool_20572893348681
// MI455X (gfx1250) — compile-verified
//
#include <hip/hip_runtime.h>

// Elementwise scale: out = x * (1/(224*224 + 1e-9)) over 102,760,448 f32.
// HBM-bound stream (822 MB single-touch, ~35 us floor at 23.3 TB/s).
// MI455X-tuned memory path:
//  - 128-bit NT global loads/stores (th:NT — zero reuse, arrays are 2x the
//    192MB L2, so keep them out of the cache hierarchy)
//  - 32-bit indexing (whole tensor < 2^31 bytes) -> GVS saddr+voffset
//    addressing, minimal SALU/VALU per access
//  - 4 independent b128 loads in flight per wave iteration (MLP)
//  - global_prefetch_b8 one chunk ahead into GL2 (gfx1250 prefetch path)
// WMMA/TDM deliberately unused: AI = 0.125 FLOP/byte; matrix units and LDS
// staging cannot help a zero-reuse stream.

typedef __attribute__((ext_vector_type(4))) float v4f;

__global__ __launch_bounds__(256) void degenerate_pool_scale_nt(
    const float* __restrict__ in, float* __restrict__ out,
    unsigned n, float scale) {
  const unsigned tid    = blockIdx.x * blockDim.x + threadIdx.x;
  const unsigned stride = gridDim.x * blockDim.x;   // in float4 units
  const unsigned n4     = n >> 2;

  const v4f* __restrict__ in4  = (const v4f*)in;
  v4f*       __restrict__ out4 = (v4f*)out;

  unsigned base = tid;

  // Steady state: all 4 chunks guaranteed in range -> branch-free body.
  for (; base + 3u * stride < n4; base += 4u * stride) {
    const unsigned next = base + 4u * stride;
    if (next < n4) {
      __builtin_prefetch((const void*)(in4 + next), 0, 0);
    }

    v4f a0 = __builtin_nontemporal_load(in4 + base);
    v4f a1 = __builtin_nontemporal_load(in4 + base + stride);
    v4f a2 = __builtin_nontemporal_load(in4 + base + 2u * stride);
    v4f a3 = __builtin_nontemporal_load(in4 + base + 3u * stride);

    a0 *= scale;
    a1 *= scale;
    a2 *= scale;
    a3 *= scale;

    __builtin_nontemporal_store(a0, out4 + base);
    __builtin_nontemporal_store(a1, out4 + base + stride);
    __builtin_nontemporal_store(a2, out4 + base + 2u * stride);
    __builtin_nontemporal_store(a3, out4 + base + 3u * stride);
  }

  // Residual float4 chunks for this thread.
  for (; base < n4; base += stride) {
    v4f a = __builtin_nontemporal_load(in4 + base);
    a *= scale;
    __builtin_nontemporal_store(a, out4 + base);
  }

  // Scalar tail for n % 4 != 0 (not hit for this shape; kept for generality).
  const unsigned r = (n4 << 2) + tid;
  if (r < n) {
    out[r] = in[r] * scale;
  }
}

extern "C" void kernel_launch(void* const* d_in, const int* in_sizes, int n_in,
                              void* d_out, int out_size, void* d_ws, size_t ws_size,
                              hipStream_t stream) {
  (void)n_in; (void)out_size; (void)d_ws; (void)ws_size;

  const float* x   = (const float*)d_in[0];
  float*       out = (float*)d_out;
  const unsigned n = (unsigned)in_sizes[0];

  // 224*224 + 1e-9 rounds to exactly 50176.0f in f32; multiply by the
  // reciprocal so the hot loop is one v_mul per component.
  const float scale = (float)(1.0 / (224.0 * 224.0 + 1e-9));

  const int threads = 256;  // 8 wave32 waves per workgroup
  const unsigned n4 = n >> 2;
  // Each thread handles 4 float4s per iteration; cap the grid so threads
  // iterate ~3x (keeps the prefetch useful, trims scheduling overhead).
  unsigned blocks = (n4 + (unsigned)threads * 4u - 1u) / ((unsigned)threads * 4u);
  if (blocks < 1u) blocks = 1u;
  if (blocks > 8192u) blocks = 8192u;

  degenerate_pool_scale_nt<<<blocks, threads, 0, stream>>>(x, out, n, scale);
}